// CRF_RNN_3977139716096
// MI455X (gfx1250) — compile-verified
//
#include <hip/hip_runtime.h>

typedef float v2f __attribute__((ext_vector_type(2)));
typedef float v4f __attribute__((ext_vector_type(4)));
typedef float v8f __attribute__((ext_vector_type(8)));

#define N_DIM 4096
#define T_DIM 32

// ---------------------------------------------------------------------------
// Kernel 1: K2[n*N+c] = sum_k W_lin[k] * kernels[(n*N+c)*8 + k]
// Streams the 512 MB kernels tensor exactly once (HBM-bound pass).
// ---------------------------------------------------------------------------
__global__ void __launch_bounds__(256) k_reduce_d(const float* __restrict__ kern,
                                                  const float* __restrict__ W_lin,
                                                  float* __restrict__ K2) {
    size_t tid = (size_t)blockIdx.x * blockDim.x + threadIdx.x; // < 16M
    const float* p = kern + tid * 8;
    v4f a = *(const v4f*)p;
    v4f b = *(const v4f*)(p + 4);
    // W_lin is wave-uniform -> scalar loads, cached
    float s = a[0] * W_lin[0] + a[1] * W_lin[1] + a[2] * W_lin[2] + a[3] * W_lin[3]
            + b[0] * W_lin[4] + b[1] * W_lin[5] + b[2] * W_lin[6] + b[3] * W_lin[7];
    K2[tid] = s;
}

// ---------------------------------------------------------------------------
// Kernel 2: L22[n] = 2 * sum_c K2[n,c]   (one block per row, K2 hot in L2)
// ---------------------------------------------------------------------------
__global__ void __launch_bounds__(256) k_row_sum(const float* __restrict__ K2,
                                                 float* __restrict__ L22) {
    __shared__ float red[256];
    const float* row = K2 + (size_t)blockIdx.x * N_DIM;
    float s = 0.f;
    for (int i = threadIdx.x; i < N_DIM; i += 256) s += row[i];
    red[threadIdx.x] = s;
    __syncthreads();
    for (int off = 128; off > 0; off >>= 1) {
        if (threadIdx.x < off) red[threadIdx.x] += red[threadIdx.x + off];
        __syncthreads();
    }
    if (threadIdx.x == 0) L22[blockIdx.x] = 2.0f * red[0];
}

// ---------------------------------------------------------------------------
// Kernel 3: orig[t,n,j] = inputs[t,n,:] @ W_feat[j,:];  x0a[t,n] = orig[t,n,0]
// ---------------------------------------------------------------------------
__global__ void __launch_bounds__(256) k_feat(const float* __restrict__ inputs,
                                              const float* __restrict__ W_feat,
                                              float* __restrict__ orig,
                                              float* __restrict__ x0) {
    size_t tid = (size_t)blockIdx.x * blockDim.x + threadIdx.x; // < 32*4096
    const float* p = inputs + tid * 8;
    v4f a = *(const v4f*)p;
    v4f b = *(const v4f*)(p + 4);
    float o[8];
#pragma unroll
    for (int j = 0; j < 8; ++j) {
        const float* w = W_feat + j * 8; // uniform -> scalar cached
        o[j] = a[0]*w[0] + a[1]*w[1] + a[2]*w[2] + a[3]*w[3]
             + b[0]*w[4] + b[1]*w[5] + b[2]*w[6] + b[3]*w[7];
    }
    v4f lo = {o[0], o[1], o[2], o[3]};
    v4f hi = {o[4], o[5], o[6], o[7]};
    *(v4f*)(orig + tid * 8)     = lo;
    *(v4f*)(orig + tid * 8 + 4) = hi;
    x0[tid] = o[0];
}

// ---------------------------------------------------------------------------
// Kernel 4: one RNN iteration.  y[t,n] = sum_c K2[n,c]*x0[t,c] via
// V_WMMA_F32_16X16X4_F32.  Block = 256 thr = 8 waves; each block owns a
// 16-column N-tile, each wave a 512-wide K-chunk, both 16-row M-tiles
// (T=32) so the B fragment (K2) is loaded once per wave.  Partial 32x16
// tiles reduced in LDS, then the rational epilogue is applied.
//
// Fragment layouts (ISA 7.12.2, f32 16x16x4):
//   A (16x4, MxK): lane l<16 -> row M=l, {v0,v1}={K,K+1}; lanes 16..31 K+2,K+3
//   B (4x16, KxN): lane l<16 -> col N=l, {v0,v1}={K,K+1}; lanes 16..31 K+2,K+3
//   C (16x16): vgpr v, lanes 0-15: (M=v,   N=lane); lanes 16-31: (M=v+8, N=lane-16)
// ---------------------------------------------------------------------------
__global__ void __launch_bounds__(256) k_rnn_step(const float* __restrict__ K2,
                                                  const float* __restrict__ x0,
                                                  const float* __restrict__ orig,
                                                  const float* __restrict__ L22,
                                                  const float* __restrict__ W_feat,
                                                  float* __restrict__ x0_next,
                                                  float* __restrict__ out,
                                                  int final_step) {
    __shared__ float lds[8][512];

    const int n0   = blockIdx.x * 16;
    const int wave = threadIdx.x >> 5;
    const int lane = threadIdx.x & 31;
    const int hf   = lane >> 4;   // half-wave selects K pair
    const int l    = lane & 15;
    const int kbase = wave * 512;

    v8f acc0 = {};  // rows t = 0..15
    v8f acc1 = {};  // rows t = 16..31

    const float* Bp  = K2 + (size_t)(n0 + l) * N_DIM + kbase + 2 * hf;
    const float* A0p = x0 + (size_t)l          * N_DIM + kbase + 2 * hf;
    const float* A1p = x0 + (size_t)(16 + l)   * N_DIM + kbase + 2 * hf;

#pragma unroll 4
    for (int kk = 0; kk < 512; kk += 4) {
        v2f b  = *(const v2f*)(Bp  + kk);
        v2f a0 = *(const v2f*)(A0p + kk);
        v2f a1 = *(const v2f*)(A1p + kk);
        acc0 = __builtin_amdgcn_wmma_f32_16x16x4_f32(false, a0, false, b,
                                                     (short)0, acc0, false, false);
        acc1 = __builtin_amdgcn_wmma_f32_16x16x4_f32(false, a1, false, b,
                                                     (short)0, acc1, false, false);
    }

    // spill fragments to LDS in (t, n) element order
#pragma unroll
    for (int v = 0; v < 8; ++v) {
        lds[wave][(v + 8 * hf) * 16 + l]        = acc0[v];
        lds[wave][(16 + v + 8 * hf) * 16 + l]   = acc1[v];
    }
    __syncthreads();

    // cross-wave K reduction + epilogue: 512 output elements, 256 threads
    for (int e = threadIdx.x; e < 512; e += 256) {
        float s = 0.f;
#pragma unroll
        for (int w = 0; w < 8; ++w) s += lds[w][e];
        const int t = e >> 4;
        const int n = n0 + (e & 15);
        const float l21  = 2.0f * s;            // layer_21[t,n]
        const float l22n = L22[n];
        const size_t tn  = (size_t)t * N_DIM + n;

        if (final_step) {
            float o[8];
#pragma unroll
            for (int j = 0; j < 8; ++j) {
                const float* w = W_feat + j * 8;
                float fw = w[0]+w[1]+w[2]+w[3]+w[4]+w[5]+w[6]+w[7];
                o[j] = (orig[tn * 8 + j] + l21) / (fw + l22n);
            }
            v4f lo = {o[0], o[1], o[2], o[3]};
            v4f hi = {o[4], o[5], o[6], o[7]};
            *(v4f*)(out + tn * 8)     = lo;
            *(v4f*)(out + tn * 8 + 4) = hi;
        } else {
            const float* w0 = W_feat;
            float fw0 = w0[0]+w0[1]+w0[2]+w0[3]+w0[4]+w0[5]+w0[6]+w0[7];
            x0_next[tn] = (orig[tn * 8] + l21) / (fw0 + l22n);
        }
    }
}

// ---------------------------------------------------------------------------
extern "C" void kernel_launch(void* const* d_in, const int* in_sizes, int n_in,
                              void* d_out, int out_size, void* d_ws, size_t ws_size,
                              hipStream_t stream) {
    const float* inputs  = (const float*)d_in[0];  // (32, 4096, 8)
    const float* kernels = (const float*)d_in[1];  // (4096, 4096, 8)
    const float* W_feat  = (const float*)d_in[2];  // (8, 8)
    const float* W_lin   = (const float*)d_in[3];  // (1, 8)
    float* out = (float*)d_out;                    // (32, 4096, 8)

    char* ws = (char*)d_ws;
    const size_t K2_BYTES   = (size_t)N_DIM * N_DIM * 4;      // 64 MB
    const size_t L22_BYTES  = (size_t)N_DIM * 4;              // 16 KB
    const size_t ORIG_BYTES = (size_t)T_DIM * N_DIM * 8 * 4;  // 4 MB
    const size_t X0_BYTES   = (size_t)T_DIM * N_DIM * 4;      // 512 KB

    float* K2   = (float*)(ws);
    float* L22  = (float*)(ws + K2_BYTES);
    float* orig = (float*)(ws + K2_BYTES + L22_BYTES);
    float* x0a  = (float*)(ws + K2_BYTES + L22_BYTES + ORIG_BYTES);
    float* x0b  = (float*)(ws + K2_BYTES + L22_BYTES + ORIG_BYTES + X0_BYTES);

    // 1) stream 512 MB kernels once -> K2 (64 MB, stays hot in 192 MB L2)
    k_reduce_d<<<(N_DIM * (N_DIM / 256)), 256, 0, stream>>>(kernels, W_lin, K2);
    // 2) layer_22 row sums
    k_row_sum<<<N_DIM, 256, 0, stream>>>(K2, L22);
    // 3) feature projection + x0 init
    k_feat<<<(T_DIM * N_DIM) / 256, 256, 0, stream>>>(inputs, W_feat, orig, x0a);
    // 4) three dependent WMMA GEMM iterations (ping-pong x0 buffers)
    k_rnn_step<<<N_DIM / 16, 256, 0, stream>>>(K2, x0a, orig, L22, W_feat, x0b, out, 0);
    k_rnn_step<<<N_DIM / 16, 256, 0, stream>>>(K2, x0b, orig, L22, W_feat, x0a, out, 0);
    k_rnn_step<<<N_DIM / 16, 256, 0, stream>>>(K2, x0a, orig, L22, W_feat, x0b, out, 1);
}